// Word2VecModel_41068477284367
// MI455X (gfx1250) — compile-verified
//
#include <hip/hip_runtime.h>
#include <math.h>

#define BATCH 4096
#define CTX 20
#define HASH 1000000
#define DIM 128
#define NSAMP 8192
#define NEG_INF_LOGIT (-1e9f)

typedef __attribute__((ext_vector_type(2))) float v2f;
typedef __attribute__((ext_vector_type(8))) float v8f;

// TF LogUniformCandidateSampler log expected count
__device__ __forceinline__ float log_expected_count(int id) {
    float idf = (float)id;
    float p = (logf(idf + 2.0f) - logf(idf + 1.0f)) / logf((float)HASH + 1.0f);
    return logf(-expm1f((float)NSAMP * log1pf(-p)));
}

// ---------------------------------------------------------------------------
// 1) h[b,:] = sum_c input_emb[input_ids[b,c], :]   (coalesced 512B row reads)
// ---------------------------------------------------------------------------
__global__ void pool_kernel(const int* __restrict__ ids,
                            const float* __restrict__ emb,
                            float* __restrict__ h) {
    const int b = blockIdx.x;
    const int d = threadIdx.x;
    const int* row = ids + b * CTX;
    float acc = 0.0f;
#pragma unroll
    for (int c = 0; c < CTX; ++c) {
        acc += emb[(long long)row[c] * DIM + d];
    }
    h[b * DIM + d] = acc;
}

// ---------------------------------------------------------------------------
// 2) sb[s] = output_bias[sid[s]] - log_expected_count(sid[s])
// ---------------------------------------------------------------------------
__global__ void sbias_kernel(const int* __restrict__ sids,
                             const float* __restrict__ bias,
                             float* __restrict__ sb) {
    int s = blockIdx.x * blockDim.x + threadIdx.x;
    if (s < NSAMP) {
        int id = sids[s];
        sb[s] = bias[id] - log_expected_count(id);
    }
}

// ---------------------------------------------------------------------------
// 3) true_logits[b] = dot(h[b], output_emb[labels[b]]) + bias - logE
//    one wave32 per row, shuffle reduce
// ---------------------------------------------------------------------------
__global__ void true_kernel(const int* __restrict__ labels,
                            const float* __restrict__ h,
                            const float* __restrict__ oemb,
                            const float* __restrict__ bias,
                            float* __restrict__ tl) {
    const int wave = threadIdx.x >> 5;
    const int lane = threadIdx.x & 31;
    const int b = blockIdx.x * 4 + wave;
    const int lab = labels[b];
    const float* hr = h + b * DIM;
    const float* wr = oemb + (long long)lab * DIM;
    float acc = 0.0f;
#pragma unroll
    for (int k = lane; k < DIM; k += 32) acc += hr[k] * wr[k];
#pragma unroll
    for (int off = 16; off; off >>= 1) acc += __shfl_xor(acc, off, 32);
    if (lane == 0) tl[b] = acc + bias[lab] - log_expected_count(lab);
}

// ---------------------------------------------------------------------------
// 4) Fused sampled-logits GEMM (fp32 WMMA 16x16x4) + online logsumexp + loss
//    Block = 16 batch rows, 8 waves split the 512 s-tiles of 16 columns.
//    Two independent accumulator fragments (even/odd K-slices) break the
//    WMMA->WMMA C-accumulation dependency chain so the matrix pipe can
//    issue back-to-back instead of stalling at full WMMA latency.
//    C fragment layout (32-bit 16x16): VGPR i holds row i (lanes 0-15) and
//    row i+8 (lanes 16-31); lane%16 selects the column within the tile.
// ---------------------------------------------------------------------------
__global__ __launch_bounds__(256) void sampled_lse_kernel(
        const float* __restrict__ h, const float* __restrict__ oemb,
        const int* __restrict__ labels, const int* __restrict__ sids,
        const float* __restrict__ sb, const float* __restrict__ tl,
        float* __restrict__ loss) {
    const int rb   = blockIdx.x;          // 16-row block of the batch
    const int wave = threadIdx.x >> 5;    // 0..7
    const int lane = threadIdx.x & 31;
    const int half = lane >> 4;           // 0: K={0,1}, 1: K={2,3}
    const int ln   = lane & 15;
    const int koff = half * 2;

    // A fragments for rows rb*16+ln, full K=128 (32 x v2f = 64 VGPRs)
    v2f af[32];
    const float* hrow = h + (rb * 16 + ln) * DIM;
#pragma unroll
    for (int kk = 0; kk < 32; ++kk) {
        af[kk].x = hrow[kk * 4 + koff];
        af[kk].y = hrow[kk * 4 + koff + 1];
    }

    // labels for this lane's 8 C-fragment rows (rows i + half*8)
    int labreg[8];
#pragma unroll
    for (int i = 0; i < 8; ++i) labreg[i] = labels[rb * 16 + i + half * 8];

    float m[8], s[8];
#pragma unroll
    for (int i = 0; i < 8; ++i) { m[i] = -INFINITY; s[i] = 0.0f; }

    // software-pipelined per-tile scalars
    int   sid_c   = sids[wave * 16 + ln];
    float sbias_c = sb[wave * 16 + ln];

    // 512 tiles of 16 columns; wave-uniform loop => EXEC all ones at WMMA
    for (int tile = wave; tile < NSAMP / 16; tile += 8) {
        const int   sid   = sid_c;
        const float sbias = sbias_c;
        const int next = tile + 8;
        if (next < NSAMP / 16) {               // uniform condition
            sid_c   = sids[next * 16 + ln];
            sbias_c = sb[next * 16 + ln];
        }
        const float* wrow = oemb + (long long)sid * DIM;

        v8f c0 = {0.f, 0.f, 0.f, 0.f, 0.f, 0.f, 0.f, 0.f};
        v8f c1 = {0.f, 0.f, 0.f, 0.f, 0.f, 0.f, 0.f, 0.f};
#pragma unroll
        for (int kk = 0; kk < 32; kk += 2) {
            v2f bf0, bf1;
            bf0.x = wrow[kk * 4 + koff];
            bf0.y = wrow[kk * 4 + koff + 1];
            bf1.x = wrow[kk * 4 + 4 + koff];
            bf1.y = wrow[kk * 4 + 4 + koff + 1];
            c0 = __builtin_amdgcn_wmma_f32_16x16x4_f32(
                     false, af[kk], false, bf0, (short)0, c0, false, false);
            c1 = __builtin_amdgcn_wmma_f32_16x16x4_f32(
                     false, af[kk + 1], false, bf1, (short)0, c1, false, false);
        }
#pragma unroll
        for (int i = 0; i < 8; ++i) {
            float l = (c0[i] + c1[i]) + sbias;
            if (sid == labreg[i]) l = NEG_INF_LOGIT;   // accidental hit
            if (l > m[i]) { s[i] = s[i] * expf(m[i] - l) + 1.0f; m[i] = l; }
            else          { s[i] += expf(l - m[i]); }
        }
    }

    // butterfly merge across the 16 columns (xor 1,2,4,8 stays in each half)
#pragma unroll
    for (int off = 1; off < 16; off <<= 1) {
#pragma unroll
        for (int i = 0; i < 8; ++i) {
            float om = __shfl_xor(m[i], off, 32);
            float os = __shfl_xor(s[i], off, 32);
            float nm = fmaxf(m[i], om);
            s[i] = s[i] * expf(m[i] - nm) + os * expf(om - nm);
            m[i] = nm;
        }
    }

    __shared__ float lm[8][16];
    __shared__ float ls[8][16];
    if (ln == 0) {
#pragma unroll
        for (int i = 0; i < 8; ++i) {
            lm[wave][i + half * 8] = m[i];
            ls[wave][i + half * 8] = s[i];
        }
    }
    __syncthreads();

    if (threadIdx.x < 16) {
        const int row = threadIdx.x;
        float M = -INFINITY, S = 0.0f;
#pragma unroll
        for (int w = 0; w < 8; ++w) {
            float om = lm[w][row], os = ls[w][row];
            float nm = fmaxf(M, om);
            S = S * expf(M - nm) + os * expf(om - nm);
            M = nm;
        }
        // fold in the true logit (column 0 of the reference concat)
        const float t = tl[rb * 16 + row];
        float nm = fmaxf(M, t);
        S = S * expf(M - nm) + expf(t - nm);
        loss[rb * 16 + row] = (nm + logf(S)) - t;
    }
}

// ---------------------------------------------------------------------------
extern "C" void kernel_launch(void* const* d_in, const int* in_sizes, int n_in,
                              void* d_out, int out_size, void* d_ws, size_t ws_size,
                              hipStream_t stream) {
    const int*   input_ids = (const int*)d_in[0];   // [4096,20] int32
    const int*   labels    = (const int*)d_in[1];   // [4096]    int32
    const int*   sids      = (const int*)d_in[2];   // [8192]    int32
    const float* iemb      = (const float*)d_in[3]; // [1e6,128] f32
    const float* oemb      = (const float*)d_in[4]; // [1e6,128] f32
    const float* obias     = (const float*)d_in[5]; // [1e6]     f32
    float* out = (float*)d_out;                     // [4096]    f32

    float* h  = (float*)d_ws;         // 4096*128
    float* sb = h + BATCH * DIM;      // 8192
    float* tl = sb + NSAMP;           // 4096

    pool_kernel<<<BATCH, DIM, 0, stream>>>(input_ids, iemb, h);
    sbias_kernel<<<NSAMP / 256, 256, 0, stream>>>(sids, obias, sb);
    true_kernel<<<BATCH / 4, 128, 0, stream>>>(labels, h, oemb, obias, tl);
    sampled_lse_kernel<<<BATCH / 16, 256, 0, stream>>>(h, oemb, labels, sids, sb, tl, out);
}